// ModuleSoftsplat_5128190951479
// MI455X (gfx1250) — compile-verified
//
#include <hip/hip_runtime.h>

#define Bn 4
#define Cn 32
#define Hn 544
#define Wn 960
#define HWn (Hn * Wn)        // 522240, divisible by 256
#define CHUNKS (HWn / 256)   // 2040 chunks per batch image

// ---------------------------------------------------------------------------
// Kernel: softmax splat. One block = 256 contiguous pixels of one batch.
// Input tile (32 channels x 256 pixels = 32 KB) is staged into LDS with the
// gfx1250 async global->LDS path, then each thread scatters its pixel's 33
// values to <=4 bilinear corners with non-returning global f32 atomics.
// ---------------------------------------------------------------------------
__global__ void __launch_bounds__(256)
splat_kernel(const float* __restrict__ in, const float* __restrict__ flow,
             const float* __restrict__ metric, float* __restrict__ out,
             float* __restrict__ norm)
{
    __shared__ float tile[Cn * 256];      // 32 KB: tile[c*256 + pix]
    const int tid   = threadIdx.x;
    const int chunk = blockIdx.x;         // 0..CHUNKS-1
    const int b     = blockIdx.y;         // 0..Bn-1
    const int base  = chunk * 256;        // pixel base within this batch image

    // LDS byte offset of tile[0]: low 32 bits of the generic pointer are the
    // workgroup-relative LDS offset, and the ptrtoint makes `tile` escape so
    // the asm's memory clobber is known to reach it.
    const unsigned tileBase = (unsigned)(uintptr_t)&tile[0];

    // --- async stage: 8 x b128 per thread covers 32 ch x 256 px x 4 B ---
    const float* inB = in + ((size_t)b * Cn) * HWn + base;
    #pragma unroll
    for (int i = 0; i < 8; ++i) {
        int linear = i * 256 + tid;            // 0..2047 (2048 x 16B = 32 KB)
        int c      = linear >> 6;              // channel 0..31
        int seg    = (linear & 63) << 4;       // byte offset inside 1 KB row
        unsigned    ldsOff = tileBase + (unsigned)(c * 1024 + seg);
        const char* gptr   = (const char*)(inB + (size_t)c * HWn) + seg;
        asm volatile("global_load_async_to_lds_b128 %0, %1, off"
                     :: "v"(ldsOff), "v"(gptr)
                     : "memory");
    }
    asm volatile("s_wait_asynccnt 0" ::: "memory");
    __syncthreads();

    // --- per-pixel flow / weights ---
    const int p = base + tid;                  // pixel index in batch image
    const int y = p / Wn;
    const int x = p - y * Wn;

    const float fx = flow[((size_t)b * 2 + 0) * HWn + p];
    const float fy = flow[((size_t)b * 2 + 1) * HWn + p];
    const float m  = __expf(metric[(size_t)b * HWn + p]);

    const float ox  = (float)x + fx;
    const float oy  = (float)y + fy;
    const float x0f = floorf(ox);
    const float y0f = floorf(oy);
    const int   x0  = (int)x0f;
    const int   y0  = (int)y0f;
    const float ax  = ox - x0f;
    const float ay  = oy - y0f;

    const float w00 = (1.0f - ax) * (1.0f - ay) * m;
    const float w01 = ax * (1.0f - ay) * m;
    const float w10 = (1.0f - ax) * ay * m;
    const float w11 = ax * ay * m;

    // Pull this pixel's 32 staged channel values into registers once
    // (conflict-free ds_load: lane stride is 4 bytes within each channel row).
    float v[Cn];
    #pragma unroll
    for (int c = 0; c < Cn; ++c) {
        v[c] = tile[c * 256 + tid];
    }

    float* outB  = out  + (size_t)b * Cn * HWn;
    float* normB = norm + (size_t)b * HWn;

    const int   cxs[4] = { x0,  x0 + 1, x0,     x0 + 1 };
    const int   cys[4] = { y0,  y0,     y0 + 1, y0 + 1 };
    const float cws[4] = { w00, w01,    w10,    w11    };

    #pragma unroll
    for (int k = 0; k < 4; ++k) {
        const int cx = cxs[k];
        const int cy = cys[k];
        if (cx < 0 || cx >= Wn || cy < 0 || cy >= Hn) continue;
        const float wm = cws[k];
        const int   q  = cy * Wn + cx;
        unsafeAtomicAdd(normB + q, wm);              // L2-resident weight plane
        #pragma unroll 8
        for (int c = 0; c < Cn; ++c) {
            unsafeAtomicAdd(outB + (size_t)c * HWn + q, v[c] * wm);
        }
    }
}

// ---------------------------------------------------------------------------
// Kernel: zero the output accumulator (d_out) and the weight plane (d_ws)
// ---------------------------------------------------------------------------
__global__ void init_zero_kernel(float4* __restrict__ out4,
                                 float4* __restrict__ norm4,
                                 int nOut4, int nNorm4) {
    int i = blockIdx.x * blockDim.x + threadIdx.x;
    float4 z = make_float4(0.0f, 0.0f, 0.0f, 0.0f);
    if (i < nOut4)  out4[i]  = z;
    if (i < nNorm4) norm4[i] = z;
}

// ---------------------------------------------------------------------------
// Kernel: normalize. One thread per pixel; one reciprocal reused across all
// 32 channels (reads/writes stay fully coalesced along W).
// ---------------------------------------------------------------------------
__global__ void normalize_kernel(float* __restrict__ out,
                                 const float* __restrict__ norm) {
    int p = blockIdx.x * blockDim.x + threadIdx.x;   // 0..Bn*HWn-1
    if (p >= Bn * HWn) return;
    const int b = p / HWn;
    const int q = p - b * HWn;
    float n = norm[p];
    n = (n == 0.0f) ? 1.0f : n;
    const float inv = 1.0f / n;
    float* o = out + (size_t)b * Cn * HWn + q;
    #pragma unroll 8
    for (int c = 0; c < Cn; ++c) {
        o[(size_t)c * HWn] *= inv;
    }
}

// ---------------------------------------------------------------------------
extern "C" void kernel_launch(void* const* d_in, const int* in_sizes, int n_in,
                              void* d_out, int out_size, void* d_ws, size_t ws_size,
                              hipStream_t stream) {
    (void)in_sizes; (void)n_in; (void)out_size; (void)ws_size;

    const float* in     = (const float*)d_in[0];   // (4,32,544,960) f32
    const float* flow   = (const float*)d_in[1];   // (4, 2,544,960) f32
    const float* metric = (const float*)d_in[2];   // (4, 1,544,960) f32
    float* out  = (float*)d_out;                   // (4,32,544,960) f32
    float* norm = (float*)d_ws;                    // (4,   544,960) f32 (8.4 MB)

    const int nOut4  = (Bn * Cn * HWn) / 4;        // 16,711,680 float4s
    const int nNorm4 = (Bn * HWn) / 4;             //    522,240 float4s

    init_zero_kernel<<<(nOut4 + 255) / 256, 256, 0, stream>>>(
        (float4*)out, (float4*)norm, nOut4, nNorm4);

    splat_kernel<<<dim3(CHUNKS, Bn), 256, 0, stream>>>(
        in, flow, metric, out, norm);

    normalize_kernel<<<(Bn * HWn + 255) / 256, 256, 0, stream>>>(out, norm);
}